// MutualInformation2_34497177321941
// MI455X (gfx1250) — compile-verified
//
#include <hip/hip_runtime.h>

#define NUM_CLASSES 5
#define NUM_BINS    256
#define IMG_B 32
#define IMG_H 512
#define IMG_W 512
#define HW    (IMG_H * IMG_W)     /* 262144 */
#define HW4   (HW / 4)            /* 65536  */
#define HIST_N (NUM_CLASSES * NUM_BINS)  /* 1280 */

typedef __attribute__((ext_vector_type(4))) float f4;
typedef __attribute__((ext_vector_type(2))) float v2f;
typedef __attribute__((ext_vector_type(8))) float v8f;

#if defined(__AMDGCN__) && __has_builtin(__builtin_amdgcn_wmma_f32_16x16x4_f32)
#define HAVE_WMMA_F32X4 1
#else
#define HAVE_WMMA_F32X4 0
#endif

// Monotone float <-> uint mapping so global min/max can use u32 atomics.
__device__ __forceinline__ unsigned encF(float f) {
    unsigned u = __float_as_uint(f);
    return (u & 0x80000000u) ? ~u : (u | 0x80000000u);
}
__device__ __forceinline__ float decF(unsigned u) {
    unsigned b = (u & 0x80000000u) ? (u ^ 0x80000000u) : ~u;
    return __uint_as_float(b);
}

// ---------------- init: ws[0..31]=min sentinel, ws[32..63]=max sentinel, hist=0
__global__ void mi_init(unsigned* __restrict__ ws) {
    int i = blockIdx.x * blockDim.x + threadIdx.x;
    if (i < 32)                ws[i] = 0xFFFFFFFFu;   // encoded +inf sentinel for min
    else if (i < 64)           ws[i] = 0u;            // encoded -inf sentinel for max
    else if (i < 64 + HIST_N)  ws[i] = 0u;            // joint histogram
}

// ---------------- pass 1: per-image min/max over 3*H*W floats
__global__ void mi_minmax(const f4* __restrict__ img4, unsigned* __restrict__ ws) {
    const int b  = blockIdx.y;
    const int N4 = 3 * HW4;   // 196608 float4 per image
    const f4* p  = img4 + (size_t)b * N4;

    float mn =  3.4e38f, mx = -3.4e38f;
    for (int i = blockIdx.x * blockDim.x + threadIdx.x; i < N4;
         i += gridDim.x * blockDim.x) {
        f4 v = p[i];
        mn = fminf(mn, fminf(fminf(v[0], v[1]), fminf(v[2], v[3])));
        mx = fmaxf(mx, fmaxf(fmaxf(v[0], v[1]), fmaxf(v[2], v[3])));
    }
    // wave32 reduction
    for (int off = 16; off; off >>= 1) {
        mn = fminf(mn, __shfl_xor(mn, off, 32));
        mx = fmaxf(mx, __shfl_xor(mx, off, 32));
    }
    __shared__ float smn[8], smx[8];
    const int wave = threadIdx.x >> 5, lane = threadIdx.x & 31;
    if (lane == 0) { smn[wave] = mn; smx[wave] = mx; }
    __syncthreads();
    if (threadIdx.x == 0) {
        const int nw = blockDim.x >> 5;
        for (int w = 1; w < nw; ++w) { mn = fminf(mn, smn[w]); mx = fmaxf(mx, smx[w]); }
        atomicMin(&ws[b],      encF(mn));
        atomicMax(&ws[32 + b], encF(mx));
    }
}

// ---------------- pass 2: gray-bin + argmax-class -> LDS histogram -> global merge
__global__ void mi_hist(const f4* __restrict__ img4, const f4* __restrict__ msk4,
                        const unsigned* __restrict__ ws, unsigned* __restrict__ ghist) {
    __shared__ unsigned lh[HIST_N];
    for (int i = threadIdx.x; i < HIST_N; i += blockDim.x) lh[i] = 0u;
    __syncthreads();

    const int b = blockIdx.y;
    const float mn  = decF(ws[b]);
    const float mx  = decF(ws[32 + b]);
    const float inv = 1.0f / (mx - mn + 1e-9f);

    const f4* r = img4 + (size_t)b * 3 * HW4;           // image[b], channels at +HW4 strides
    const f4* m = msk4 + (size_t)b * NUM_CLASSES * HW4; // mask[b]

    for (int i = blockIdx.x * blockDim.x + threadIdx.x; i < HW4;
         i += gridDim.x * blockDim.x) {
        f4 vr = r[i];
        f4 vg = r[i + HW4];
        f4 vb = r[i + 2 * HW4];
        // mask is single-use: non-temporal so the image stays resident in L2
        f4 m0 = __builtin_nontemporal_load(&m[i]);
        f4 m1 = __builtin_nontemporal_load(&m[i + HW4]);
        f4 m2 = __builtin_nontemporal_load(&m[i + 2 * HW4]);
        f4 m3 = __builtin_nontemporal_load(&m[i + 3 * HW4]);
        f4 m4 = __builtin_nontemporal_load(&m[i + 4 * HW4]);
#pragma unroll
        for (int j = 0; j < 4; ++j) {
            float gray = 0.2989f * ((vr[j] - mn) * inv)
                       + 0.587f  * ((vg[j] - mn) * inv)
                       + 0.114f  * ((vb[j] - mn) * inv);
            int bin = (int)floorf(gray * 255.0f);
            bin = bin < 0 ? 0 : (bin > (NUM_BINS - 1) ? (NUM_BINS - 1) : bin);
            // first-max argmax (strict >) matches jnp.argmax tie-breaking
            float bv = m0[j]; int cls = 0;
            if (m1[j] > bv) { bv = m1[j]; cls = 1; }
            if (m2[j] > bv) { bv = m2[j]; cls = 2; }
            if (m3[j] > bv) { bv = m3[j]; cls = 3; }
            if (m4[j] > bv) { bv = m4[j]; cls = 4; }
            atomicAdd(&lh[cls * NUM_BINS + bin], 1u);
        }
    }
    __syncthreads();
    for (int i = threadIdx.x; i < HIST_N; i += blockDim.x) {
        unsigned v = lh[i];
        if (v) atomicAdd(&ghist[i], v);
    }
}

// ---------------- pass 3: finalize (single block). Class-sum via V_WMMA_F32_16X16X4_F32.
__global__ void mi_finalize(const unsigned* __restrict__ ghist, float* __restrict__ out) {
    __shared__ float pj[HIST_N];
    __shared__ float pimg[NUM_BINS];
    __shared__ float pmask[NUM_CLASSES];
    __shared__ float rzs;

    const float invTotal = 1.0f / ((float)IMG_B * (float)IMG_H * (float)IMG_W);

    for (int i = threadIdx.x; i < HIST_N; i += blockDim.x) pj[i] = (float)ghist[i];
    __syncthreads();
    if (threadIdx.x == 0) {
        float rz = 0.0f;
        for (int c = 0; c < NUM_CLASSES; ++c) rz += pj[c * NUM_BINS];
        rzs = rz;  // total zero-bin pixel count (counts < 2^24 -> exact in f32)
    }
    __syncthreads();
    for (int i = threadIdx.x; i < HIST_N; i += blockDim.x) {
        float cnt = ((i & (NUM_BINS - 1)) == 0) ? rzs : pj[i];
        pj[i] = cnt * invTotal;   // p_joint
    }
    __syncthreads();

    // p_img[n] = sum_c p_joint[c][n]
#if HAVE_WMMA_F32X4
    if (threadIdx.x < 32) {   // wave 0 only: EXEC all ones within the wave
        const int lane = threadIdx.x;
        const int mrow = lane & 15;
        const bool hi  = lane >= 16;
        v2f bones; bones[0] = 1.0f; bones[1] = 1.0f;   // B = ones (layout-independent)
        for (int ch = 0; ch < 16; ++ch) {
            const int n0 = ch * 16;
            // A (16x4 f32): VGPR0 = K0 (lanes 0-15) / K2 (lanes 16-31); VGPR1 = K1 / K3
            v2f a1; // classes 0..3
            a1[0] = pj[(hi ? 2 : 0) * NUM_BINS + n0 + mrow];
            a1[1] = pj[(hi ? 3 : 1) * NUM_BINS + n0 + mrow];
            v2f a2; // class 4 in K0, rest zero
            a2[0] = hi ? 0.0f : pj[4 * NUM_BINS + n0 + mrow];
            a2[1] = 0.0f;
            v8f acc = {0.0f, 0.0f, 0.0f, 0.0f, 0.0f, 0.0f, 0.0f, 0.0f};
            acc = __builtin_amdgcn_wmma_f32_16x16x4_f32(false, a1, false, bones,
                                                        (short)0, acc, false, false);
            acc = __builtin_amdgcn_wmma_f32_16x16x4_f32(false, a2, false, bones,
                                                        (short)0, acc, false, false);
            // D layout: VGPR j, lanes 0-15 -> M=j,N=lane ; lanes 16-31 -> M=j+8
            if (lane == 0)       { for (int j = 0; j < 8; ++j) pimg[n0 + j]     = acc[j]; }
            else if (lane == 16) { for (int j = 0; j < 8; ++j) pimg[n0 + 8 + j] = acc[j]; }
        }
    }
#else
    for (int n = threadIdx.x; n < NUM_BINS; n += blockDim.x) {
        float s = 0.0f;
        for (int c = 0; c < NUM_CLASSES; ++c) s += pj[c * NUM_BINS + n];
        pimg[n] = s;
    }
#endif
    __syncthreads();
    if (threadIdx.x < NUM_CLASSES) {
        float s = 0.0f;
        for (int n = 0; n < NUM_BINS; ++n) s += pj[threadIdx.x * NUM_BINS + n];
        pmask[threadIdx.x] = s;
    }
    __syncthreads();

    // outputs: p_joint[1280] | p_img[256] | p_mask[5] | p_img_mask[1280] | p_mask_img[1280]
    for (int i = threadIdx.x; i < NUM_BINS; i += blockDim.x) out[1280 + i] = pimg[i];
    if (threadIdx.x < NUM_CLASSES) out[1536 + threadIdx.x] = pmask[threadIdx.x];
    for (int i = threadIdx.x; i < HIST_N; i += blockDim.x) {
        const int c = i >> 8, n = i & (NUM_BINS - 1);
        const float v = pj[i];
        out[i]        = v;                           // p_joint
        out[1541 + i] = v / (pmask[c] + 1e-9f);      // p_img_mask
        out[2821 + i] = v / (pimg[n]  + 1e-9f);      // p_mask_img
    }
}

extern "C" void kernel_launch(void* const* d_in, const int* in_sizes, int n_in,
                              void* d_out, int out_size, void* d_ws, size_t ws_size,
                              hipStream_t stream) {
    const f4* img4 = (const f4*)d_in[0];   // image  [32,3,512,512] f32
    const f4* msk4 = (const f4*)d_in[1];   // mask   [32,5,512,512] f32
    unsigned* ws    = (unsigned*)d_ws;     // [0..31] minEnc, [32..63] maxEnc, [64..) hist
    unsigned* ghist = ws + 64;
    float* out = (float*)d_out;

    mi_init    <<<6, 256, 0, stream>>>(ws);
    mi_minmax  <<<dim3(128, IMG_B), 256, 0, stream>>>(img4, ws);
    mi_hist    <<<dim3(64,  IMG_B), 256, 0, stream>>>(img4, msk4, ws, ghist);
    mi_finalize<<<1, 256, 0, stream>>>(ghist, out);
}